// ClassificationPointTransformer_57200374448395
// MI455X (gfx1250) — compile-verified
//
#include <hip/hip_runtime.h>
#include <hip/hip_bf16.h>

typedef __attribute__((ext_vector_type(2))) float v2f;
typedef __attribute__((ext_vector_type(8))) float v8f;

#define B_   16
#define KNN_ 16

// ---------------------------------------------------------------------------
// fp32 WMMA GEMM (V_WMMA_F32_16X16X4_F32), wave32, one wave per block.
//
// VGPR layouts (CDNA5 ISA 7.12.2, wave32):
//   A 16x4 : lane l holds A[l&15, 2*(l>>4)+j] in a[j]
//   B 4x16 : lane l holds B[2*(l>>4)+j, l&15] in b[j]
//   C 16x16: lane l holds C[v + 8*(l>>4), l&15] in acc[v]
//
// Fast kernel: requires Kd%4==0 and N%32==0. Each wave computes a 16x32
// strip (two accumulators share one A fragment). All guards are branchless
// (clamped addresses + multiply-by-mask), so no EXEC save/restore in the
// K-loop and EXEC stays all-ones for WMMA.
// ---------------------------------------------------------------------------
template<bool BIAS, bool RELU>
__global__ __launch_bounds__(32)
void gemm_fast_kernel(const float* __restrict__ X, const float* __restrict__ W,
                      const float* __restrict__ bias, float* __restrict__ Y,
                      int M, int N, int Kd)
{
    const int tm   = blockIdx.x;
    const int tn   = blockIdx.y;
    const int lane = (int)threadIdx.x;
    const int half = lane >> 4;
    const int l16  = lane & 15;

    const int rowbase = tm * 16;
    const int col0 = tn * 32 + l16;
    const int col1 = col0 + 16;
    const int arow = rowbase + l16;
    const int ar   = (arow < M) ? arow : (M - 1);       // clamped: load always legal
    const float amask = (arow < M) ? 1.0f : 0.0f;       // branchless row guard

    const float* Xr = X + (size_t)ar * Kd + 2 * half;

    v8f acc0 = {};
    v8f acc1 = {};
#pragma unroll 2
    for (int k0 = 0; k0 < Kd; k0 += 4) {
        v2f a = *(const v2f*)(Xr + k0);                 // global_load_b64
        a[0] *= amask;
        a[1] *= amask;
        const int kr = k0 + 2 * half;
        v2f b0, b1;
        b0[0] = W[(size_t)kr * N + col0];
        b0[1] = W[(size_t)(kr + 1) * N + col0];
        b1[0] = W[(size_t)kr * N + col1];
        b1[1] = W[(size_t)(kr + 1) * N + col1];
        if (k0 + 16 < Kd)
            __builtin_prefetch(Xr + k0 + 16, 0, 1);     // global_prefetch next X line
        // (neg_a, A, neg_b, B, c_mod, C, reuse_a, reuse_b)
        acc0 = __builtin_amdgcn_wmma_f32_16x16x4_f32(false, a, false, b0,
                                                     (short)0, acc0, false, false);
        acc1 = __builtin_amdgcn_wmma_f32_16x16x4_f32(false, a, false, b1,
                                                     (short)0, acc1, false, false);
    }

    const int rbase = rowbase + half * 8;
    if (rowbase + 16 <= M) {                            // uniform branch: interior tile
#pragma unroll
        for (int v = 0; v < 8; ++v) {
            const int r = rbase + v;
            float v0 = acc0[v], v1 = acc1[v];
            if (BIAS) { v0 += bias[col0]; v1 += bias[col1]; }
            if (RELU) { v0 = fmaxf(v0, 0.0f); v1 = fmaxf(v1, 0.0f); }
            Y[(size_t)r * N + col0] = v0;
            Y[(size_t)r * N + col1] = v1;
        }
    } else {                                            // M-edge tile: guarded stores
#pragma unroll
        for (int v = 0; v < 8; ++v) {
            const int r = rbase + v;
            if (r < M) {
                float v0 = acc0[v], v1 = acc1[v];
                if (BIAS) { v0 += bias[col0]; v1 += bias[col1]; }
                if (RELU) { v0 = fmaxf(v0, 0.0f); v1 = fmaxf(v1, 0.0f); }
                Y[(size_t)r * N + col0] = v0;
                Y[(size_t)r * N + col1] = v1;
            }
        }
    }
}

// Generic kernel: any M/N/Kd (used for Kd=3 layers and the N=2 head).
// Guards are branchless: clamped addresses, unconditional loads, v_cndmask.
template<bool BIAS, bool RELU>
__global__ __launch_bounds__(32)
void gemm_any_kernel(const float* __restrict__ X, const float* __restrict__ W,
                     const float* __restrict__ bias, float* __restrict__ Y,
                     int M, int N, int Kd)
{
    const int tm   = blockIdx.x;
    const int tn   = blockIdx.y;
    const int lane = (int)threadIdx.x;
    const int half = lane >> 4;
    const int l16  = lane & 15;

    const int arow = tm * 16 + l16;
    const int bcol = tn * 16 + l16;
    const int ar = (arow < M) ? arow : (M - 1);
    const int bc = (bcol < N) ? bcol : (N - 1);
    const float amask = (arow < M) ? 1.0f : 0.0f;
    const float bmask = (bcol < N) ? 1.0f : 0.0f;

    v8f acc = {};
    for (int k0 = 0; k0 < Kd; k0 += 4) {
        v2f a, b;
#pragma unroll
        for (int j = 0; j < 2; ++j) {
            const int k  = k0 + half * 2 + j;
            const int kc = (k < Kd) ? k : (Kd - 1);
            const float av = X[(size_t)ar * Kd + kc];   // always in-bounds
            const float bv = W[(size_t)kc * N + bc];
            a[j] = (k < Kd) ? av * amask : 0.0f;        // v_cndmask, no branch
            b[j] = (k < Kd) ? bv * bmask : 0.0f;
        }
        acc = __builtin_amdgcn_wmma_f32_16x16x4_f32(false, a, false, b,
                                                    (short)0, acc, false, false);
    }
#pragma unroll
    for (int v = 0; v < 8; ++v) {
        const int r = tm * 16 + half * 8 + v;
        if (r < M && bcol < N) {
            float val = acc[v];
            if (BIAS) val += bias[bcol];
            if (RELU) val = fmaxf(val, 0.0f);
            Y[(size_t)r * N + bcol] = val;
        }
    }
}

static inline void gemm(const float* X, const float* W, const float* bias, float* Y,
                        int M, int N, int Kd, bool useBias, bool relu, hipStream_t s)
{
    if ((Kd % 4 == 0) && (N % 32 == 0)) {
        dim3 g((M + 15) / 16, N / 32);
        if (useBias && relu)
            gemm_fast_kernel<true, true><<<g, dim3(32), 0, s>>>(X, W, bias, Y, M, N, Kd);
        else if (useBias)
            gemm_fast_kernel<true, false><<<g, dim3(32), 0, s>>>(X, W, bias, Y, M, N, Kd);
        else
            gemm_fast_kernel<false, false><<<g, dim3(32), 0, s>>>(X, W, bias, Y, M, N, Kd);
    } else {
        dim3 g((M + 15) / 16, (N + 15) / 16);
        if (useBias && relu)
            gemm_any_kernel<true, true><<<g, dim3(32), 0, s>>>(X, W, bias, Y, M, N, Kd);
        else if (useBias)
            gemm_any_kernel<true, false><<<g, dim3(32), 0, s>>>(X, W, bias, Y, M, N, Kd);
        else
            gemm_any_kernel<false, false><<<g, dim3(32), 0, s>>>(X, W, bias, Y, M, N, Kd);
    }
}

// ---------------------------------------------------------------------------
// kNN: one thread per query, 16-entry insertion list. Strict '<' comparisons
// reproduce top_k / argmax stable tie-breaking (lowest index wins).
// blockIdx.y = batch.  Unfilled slots (n too small) fall back to self index.
// ---------------------------------------------------------------------------
__global__ void knn_kernel(const float* __restrict__ qpos, const float* __restrict__ ppos,
                           int nq, int npt, int* __restrict__ nbr, int stride,
                           int excludeSelf, int appendSelf)
{
    const int q = blockIdx.x * blockDim.x + threadIdx.x;
    const int b = blockIdx.y;
    if (q >= nq) return;
    const float* qp = qpos + (size_t)b * nq * 3;
    const float* pp = ppos + (size_t)b * npt * 3;
    const float qx = qp[q * 3], qy = qp[q * 3 + 1], qz = qp[q * 3 + 2];

    float bd[KNN_]; int bi[KNN_];
    for (int t = 0; t < KNN_; ++t) { bd[t] = 3.402823e38f; bi[t] = q; }

    for (int j = 0; j < npt; ++j) {
        if (excludeSelf && j == q) continue;
        const float dx = qx - pp[j * 3];
        const float dy = qy - pp[j * 3 + 1];
        const float dz = qz - pp[j * 3 + 2];
        const float d = dx * dx + dy * dy + dz * dz;
        if (d < bd[KNN_ - 1]) {
            int t = KNN_ - 1;
            while (t > 0 && bd[t - 1] > d) { bd[t] = bd[t - 1]; bi[t] = bi[t - 1]; --t; }
            bd[t] = d; bi[t] = j;
        }
    }
    int* row = nbr + (size_t)b * nq * stride + (size_t)q * stride;
    for (int t = 0; t < KNN_; ++t) row[t] = bi[t];
    if (appendSelf) row[KNN_] = q;
}

// ---------------------------------------------------------------------------
// Farthest point sampling, one workgroup per batch, dist array in LDS.
// Matches reference: idx[0]=0, then argmax (lowest index on ties).
// ---------------------------------------------------------------------------
__global__ __launch_bounds__(256)
void fps_kernel(const float* __restrict__ posIn, int n, int nsub, int* __restrict__ outIdx)
{
    const int b = blockIdx.x;
    const float* pos = posIn + (size_t)b * n * 3;
    int* oidx = outIdx + (size_t)b * nsub;

    __shared__ float dist[2048];
    __shared__ float rv[256];
    __shared__ int   ri[256];
    const int t = threadIdx.x;

    const float p0x = pos[0], p0y = pos[1], p0z = pos[2];
    for (int j = t; j < n; j += 256) {
        const float dx = pos[j * 3] - p0x, dy = pos[j * 3 + 1] - p0y, dz = pos[j * 3 + 2] - p0z;
        dist[j] = dx * dx + dy * dy + dz * dz;
    }
    if (t == 0) oidx[0] = 0;
    __syncthreads();

    for (int i = 1; i < nsub; ++i) {
        float bv = -1.0f; int bj = 0x7fffffff;
        for (int j = t; j < n; j += 256) {
            const float v = dist[j];
            if (v > bv) { bv = v; bj = j; }
        }
        rv[t] = bv; ri[t] = bj;
        __syncthreads();
        for (int s = 128; s > 0; s >>= 1) {
            if (t < s) {
                if (rv[t + s] > rv[t] || (rv[t + s] == rv[t] && ri[t + s] < ri[t])) {
                    rv[t] = rv[t + s]; ri[t] = ri[t + s];
                }
            }
            __syncthreads();
        }
        const int nxt = ri[0];
        if (t == 0) oidx[i] = nxt;
        const float nx = pos[nxt * 3], ny = pos[nxt * 3 + 1], nz = pos[nxt * 3 + 2];
        for (int j = t; j < n; j += 256) {
            const float dx = pos[j * 3] - nx, dy = pos[j * 3 + 1] - ny, dz = pos[j * 3 + 2] - nz;
            const float d = dx * dx + dy * dy + dz * dz;
            if (d < dist[j]) dist[j] = d;
        }
        __syncthreads();
    }
}

__global__ void gather_pos_kernel(const float* __restrict__ pos, const int* __restrict__ idx,
                                  float* __restrict__ outp, int nsub, int n)
{
    const int i = blockIdx.x * blockDim.x + threadIdx.x;
    const int b = blockIdx.y;
    if (i >= nsub * 3) return;
    const int pt = i / 3, d = i % 3;
    outp[(size_t)b * nsub * 3 + i] =
        pos[(size_t)b * n * 3 + (size_t)idx[(size_t)b * nsub + pt] * 3 + d];
}

// pdiff[e,3] = pos[i] - pos[nbr[e]]   (per-batch pointers, e = i*17+k)
__global__ void pdiff_kernel(const float* __restrict__ pos, const int* __restrict__ nbr,
                             float* __restrict__ pdiff, int n)
{
    const int e = blockIdx.x * blockDim.x + threadIdx.x;
    if (e >= n * 17) return;
    const int i = e / 17;
    const int j = nbr[e];
    pdiff[e * 3 + 0] = pos[i * 3 + 0] - pos[j * 3 + 0];
    pdiff[e * 3 + 1] = pos[i * 3 + 1] - pos[j * 3 + 1];
    pdiff[e * 3 + 2] = pos[i * 3 + 2] - pos[j * 3 + 2];
}

// ain[e,c] = a_dst[i,c] - a_src[nbr[e],c] + delta[e,c]
__global__ void ain_kernel(const float* __restrict__ a_src, const float* __restrict__ a_dst,
                           const int* __restrict__ nbr, const float* __restrict__ delta,
                           float* __restrict__ ain, int n, int c)
{
    const int idx = blockIdx.x * blockDim.x + threadIdx.x;
    if (idx >= n * 17 * c) return;
    const int ch = idx % c;
    const int e  = idx / c;
    const int i  = e / 17;
    const int j  = nbr[e];
    ain[idx] = a_dst[(size_t)i * c + ch] - a_src[(size_t)j * c + ch] + delta[idx];
}

// channelwise softmax over 17 neighbors, then out[i,c] = sum alpha*(h[nbr]+delta)
__global__ void softagg_kernel(const float* __restrict__ alpha, const float* __restrict__ delta,
                               const float* __restrict__ hlin, const int* __restrict__ nbr,
                               float* __restrict__ outp, int n, int c)
{
    const int idx = blockIdx.x * blockDim.x + threadIdx.x;
    if (idx >= n * c) return;
    const int i = idx / c, ch = idx % c;
    float m = -3.402823e38f;
    for (int k = 0; k < 17; ++k)
        m = fmaxf(m, alpha[(size_t)(i * 17 + k) * c + ch]);
    float ssum = 0.0f, acc = 0.0f;
    for (int k = 0; k < 17; ++k) {
        const float e = __expf(alpha[(size_t)(i * 17 + k) * c + ch] - m);
        ssum += e;
        const int j = nbr[i * 17 + k];
        acc += e * (hlin[(size_t)j * c + ch] + delta[(size_t)(i * 17 + k) * c + ch]);
    }
    outp[idx] = acc / ssum;
}

// neighbor max pool (t_down), blockIdx.y = batch
__global__ void maxpool_kernel(const float* __restrict__ hin, const int* __restrict__ nbr,
                               float* __restrict__ outp, int nsub, int c, int nprev)
{
    const int idx = blockIdx.x * blockDim.x + threadIdx.x;
    const int b = blockIdx.y;
    if (idx >= nsub * c) return;
    const int i = idx / c, ch = idx % c;
    const float* hb = hin + (size_t)b * nprev * c;
    const int* nb = nbr + (size_t)b * nsub * KNN_ + (size_t)i * KNN_;
    float m = -3.402823e38f;
#pragma unroll 4
    for (int k = 0; k < KNN_; ++k)
        m = fmaxf(m, hb[(size_t)nb[k] * c + ch]);
    outp[(size_t)b * nsub * c + idx] = m;
}

__global__ void meanpool_kernel(const float* __restrict__ hin, float* __restrict__ g, int n, int c)
{
    const int idx = blockIdx.x * blockDim.x + threadIdx.x;
    if (idx >= B_ * c) return;
    const int b = idx / c, ch = idx % c;
    float s = 0.0f;
    for (int i = 0; i < n; ++i) s += hin[((size_t)b * n + i) * c + ch];
    g[idx] = s / (float)n;
}

__global__ void softmax2_kernel(const float* __restrict__ logits, float* __restrict__ outp)
{
    const int b = blockIdx.x * blockDim.x + threadIdx.x;
    if (b >= B_) return;
    const float a = logits[b * 2], c2 = logits[b * 2 + 1];
    const float m = fmaxf(a, c2);
    const float ea = __expf(a - m), eb = __expf(c2 - m);
    const float inv = 1.0f / (ea + eb);
    outp[b * 2]     = ea * inv;
    outp[b * 2 + 1] = eb * inv;
}

// ---------------------------------------------------------------------------
// Host-side orchestration
// ---------------------------------------------------------------------------
struct BlockP {
    const float *liw, *lib, *low, *lob;   // lin_in / lin_out
    const float *wl, *ws, *wd;            // w_lin / w_src / w_dst
    const float *p0w, *p0b, *p1w, *p1b;   // pos_nn
    const float *a0w, *a0b, *a1w, *a1b;   // attn_nn
};

static BlockP loadBlock(void* const* d_in, int base)
{
    BlockP p;
    auto F = [&](int i) { return (const float*)d_in[i]; };
    p.liw = F(base + 0);  p.lib = F(base + 1);
    p.low = F(base + 2);  p.lob = F(base + 3);
    p.wl  = F(base + 4);  p.ws  = F(base + 5);  p.wd = F(base + 6);
    p.p0w = F(base + 7);  p.p0b = F(base + 8);
    p.p1w = F(base + 9);  p.p1b = F(base + 10);
    p.a0w = F(base + 11); p.a0b = F(base + 12);
    p.a1w = F(base + 13); p.a1b = F(base + 14);
    return p;
}

static void runBlock(const BlockP& p, const float* posCur, int n, int c,
                     const float* h_in, float* h_out,
                     float* t_lin, float* h_lin, float* a_src, float* a_dst,
                     int* nbr, float* pdiff, float* hid, float* delta,
                     float* ainb, float* alpha, hipStream_t s)
{
    const int rows = B_ * n;
    const int R = n * 17;

    // x = relu(lin_in(x))
    gemm(h_in, p.liw, p.lib, t_lin, rows, c, c, true, true, s);
    // knn graph with self-loop appended (stride 17)
    knn_kernel<<<dim3((n + 127) / 128, B_), dim3(128), 0, s>>>(
        posCur, posCur, n, n, nbr, 17, 1, 1);
    // h = x@w_lin ; a_src = x@w_src ; a_dst = x@w_dst
    gemm(t_lin, p.wl, nullptr, h_lin, rows, c, c, false, false, s);
    gemm(t_lin, p.ws, nullptr, a_src, rows, c, c, false, false, s);
    gemm(t_lin, p.wd, nullptr, a_dst, rows, c, c, false, false, s);

    for (int b = 0; b < B_; ++b) {
        const float* posb = posCur + (size_t)b * n * 3;
        const int* nbrb = nbr + (size_t)b * R;
        pdiff_kernel<<<dim3((R + 255) / 256), dim3(256), 0, s>>>(posb, nbrb, pdiff, n);
        // delta = relu(relu(pdiff @ p0 + b0) @ p1 + b1)
        gemm(pdiff, p.p0w, p.p0b, hid, R, 64, 3, true, true, s);
        gemm(hid, p.p1w, p.p1b, delta, R, c, 64, true, true, s);
        // alpha input = a_dst[i] - a_src[j] + delta
        ain_kernel<<<dim3((R * c + 255) / 256), dim3(256), 0, s>>>(
            a_src + (size_t)b * n * c, a_dst + (size_t)b * n * c, nbrb, delta, ainb, n, c);
        gemm(ainb, p.a0w, p.a0b, hid, R, 64, c, true, true, s);
        gemm(hid, p.a1w, p.a1b, alpha, R, c, 64, true, true, s);
        // softmax over neighbors + aggregate into t_lin (conv output, reuses buffer)
        softagg_kernel<<<dim3((n * c + 255) / 256), dim3(256), 0, s>>>(
            alpha, delta, h_lin + (size_t)b * n * c, nbrb, t_lin + (size_t)b * n * c, n, c);
    }
    // x = relu(lin_out(conv))
    gemm(t_lin, p.low, p.lob, h_out, rows, c, c, true, true, s);
}

static void runDown(const float* tdw, const float* tdb,
                    const float* posCur, float* posNext,
                    const float* h_in, float* h_next, float* t_lin,
                    int n, int nsub, int cin, int cout,
                    int* fpsIdx, int* nbrP, hipStream_t s)
{
    fps_kernel<<<dim3(B_), dim3(256), 0, s>>>(posCur, n, nsub, fpsIdx);
    gather_pos_kernel<<<dim3((nsub * 3 + 127) / 128, B_), dim3(128), 0, s>>>(
        posCur, fpsIdx, posNext, nsub, n);
    knn_kernel<<<dim3((nsub + 127) / 128, B_), dim3(128), 0, s>>>(
        posNext, posCur, nsub, n, nbrP, KNN_, 0, 0);
    gemm(h_in, tdw, tdb, t_lin, B_ * n, cout, cin, true, true, s);
    maxpool_kernel<<<dim3((nsub * cout + 255) / 256, B_), dim3(256), 0, s>>>(
        t_lin, nbrP, h_next, nsub, cout, n);
}

extern "C" void kernel_launch(void* const* d_in, const int* in_sizes, int n_in,
                              void* d_out, int out_size, void* d_ws, size_t ws_size,
                              hipStream_t stream)
{
    (void)in_sizes; (void)n_in; (void)out_size; (void)ws_size;
    auto F = [&](int i) { return (const float*)d_in[i]; };

    // --- input flattening (setup_inputs insertion order) ---
    const float* x    = F(0);   // [16,2048,3]
    const float* pos0 = F(1);   // [16,2048,3]
    const float* mw = F(2); const float* mb = F(3);
    BlockP block0 = loadBlock(d_in, 4);
    const float* tdw[4]; const float* tdb[4];
    for (int i = 0; i < 4; ++i) { tdw[i] = F(19 + 2 * i); tdb[i] = F(20 + 2 * i); }
    BlockP blocks[4];
    for (int i = 0; i < 4; ++i) blocks[i] = loadBlock(d_in, 27 + 15 * i);
    const float* o1w = F(87); const float* o1b = F(88);
    const float* o2w = F(89); const float* o2b = F(90);

    // --- workspace carve ---
    char* base = (char*)d_ws;
    size_t ofs = 0;
    auto AF = [&](size_t nf) { float* p = (float*)(base + ofs);
                               ofs += ((nf * sizeof(float) + 255) & ~(size_t)255); return p; };
    auto AI = [&](size_t ni) { int* p = (int*)(base + ofs);
                               ofs += ((ni * sizeof(int) + 255) & ~(size_t)255); return p; };

    float* hA    = AF((size_t)B_ * 2048 * 32);   // current features (block outputs)
    float* hB    = AF((size_t)B_ * 512 * 64);    // pooled features
    float* t_lin = AF((size_t)B_ * 2048 * 64);   // largest linear intermediate
    float* h_lin = AF((size_t)B_ * 2048 * 32);
    float* a_src = AF((size_t)B_ * 2048 * 32);
    float* a_dst = AF((size_t)B_ * 2048 * 32);
    float* posWA = AF((size_t)B_ * 512 * 3);
    float* posWB = AF((size_t)B_ * 512 * 3);
    float* pdiff = AF((size_t)2048 * 17 * 3);    // per-batch pair scratch
    float* hid   = AF((size_t)2048 * 17 * 64);
    float* delta = AF((size_t)2048 * 17 * 32);
    float* ainb  = AF((size_t)2048 * 17 * 32);
    float* alpha = AF((size_t)2048 * 17 * 32);
    float* g     = AF((size_t)B_ * 512);
    float* t1    = AF((size_t)B_ * 64);
    float* logit = AF((size_t)B_ * 2);
    int* nbr     = AI((size_t)B_ * 2048 * 17);
    int* nbrP    = AI((size_t)B_ * 512 * KNN_);
    int* fpsIdx  = AI((size_t)B_ * 512);

    const int dims[5] = {32, 64, 128, 256, 512};
    const int ns[5]   = {2048, 512, 128, 32, 8};

    // h = relu(mlp_input(x))   (Kd=3 -> generic kernel)
    gemm(x, mw, mb, hA, B_ * 2048, 32, 3, true, true, stream);
    // block0 at full resolution
    runBlock(block0, pos0, 2048, 32, hA, hA,
             t_lin, h_lin, a_src, a_dst, nbr, pdiff, hid, delta, ainb, alpha, stream);

    const float* posCur = pos0;
    float* posBufs[2] = {posWA, posWB};
    for (int L = 0; L < 4; ++L) {
        const int n = ns[L], nsub = ns[L + 1];
        const int cin = dims[L], cout = dims[L + 1];
        float* posNext = posBufs[L & 1];
        runDown(tdw[L], tdb[L], posCur, posNext, hA, hB, t_lin,
                n, nsub, cin, cout, fpsIdx, nbrP, stream);
        runBlock(blocks[L], posNext, nsub, cout, hB, hA,
                 t_lin, h_lin, a_src, a_dst, nbr, pdiff, hid, delta, ainb, alpha, stream);
        posCur = posNext;
    }

    // head: mean-pool -> lin(512,64)+relu -> lin(64,2) -> softmax
    meanpool_kernel<<<dim3((B_ * 512 + 255) / 256), dim3(256), 0, stream>>>(hA, g, 8, 512);
    gemm(g, o1w, o1b, t1, B_, 64, 512, true, true, stream);
    gemm(t1, o2w, o2b, logit, B_, 2, 64, true, false, stream);
    softmax2_kernel<<<dim3(1), dim3(32), 0, stream>>>(logit, (float*)d_out);
}